// Agent_74174085202347
// MI455X (gfx1250) — compile-verified
//
#include <hip/hip_runtime.h>
#include <math.h>

#define N_NODES 20000
#define N_EDGES 640000
#define HIDC    128
#define OUTD    64

typedef __attribute__((ext_vector_type(2))) float v2f;
typedef __attribute__((ext_vector_type(8))) float v8f;

// ---------------------------------------------------------------------------
// WMMA wrapper: D(16x16 f32) += A(16x4 f32) * B(4x16 f32)
// A frag: all lanes hold M=lane&15; VGPR0=K{0,2}, VGPR1=K{1,3} (sel by lane>>4)
// B frag: VGPR0 = rows K{0,2} striped over lanes (N=lane&15), VGPR1 = K{1,3}
// C/D  : VGPR r -> M = r + 8*(lane>>4), N = lane&15
// ---------------------------------------------------------------------------
__device__ __forceinline__ v8f wmma_f32(v2f a, v2f b, v8f c) {
    return __builtin_amdgcn_wmma_f32_16x16x4_f32(
        /*neg_a=*/false, a, /*neg_b=*/false, b,
        /*c_mod=*/(short)0, c, /*reuse_a=*/false, /*reuse_b=*/false);
}

template <int ACT>
__device__ __forceinline__ float apply_act(float x) {
    if (ACT == 1) return fmaxf(x, 0.0f);                                    // ReLU
    if (ACT == 2) return 0.5f * x * (1.0f + erff(x * 0.7071067811865476f)); // exact GELU
    return x;
}

// ---------------------------------------------------------------------------
// Tiled WMMA GEMM: C[N x OUTC] = act( A0*rs0 @ W0 (+ A1 @ W1) + bias )
// K = HIDC = 128. Block covers MTILES*16 rows; one wave per 16-col slice
// computes MTILES 16x16 C tiles, reusing each register-held B fragment
// MTILES times (4x fewer B loads per WMMA than one-tile-per-wave).
// A staged in LDS with 132-float row stride (conflict-free across 64 banks).
// ---------------------------------------------------------------------------
template <int OUTC, int ACT, bool DUAL, int MTILES>
__global__ void __launch_bounds__(32 * (OUTC / 16))
gemm_wmma(const float* __restrict__ A0, const float* __restrict__ rs0,
          const float* __restrict__ W0,
          const float* __restrict__ A1, const float* __restrict__ W1w,
          const float* __restrict__ bias, float* __restrict__ C)
{
    constexpr int NT  = 32 * (OUTC / 16);
    constexpr int MT  = 16 * MTILES;
    constexpr int LDA = HIDC + 4;            // padded LDS row stride
    __shared__ __attribute__((aligned(16))) float As0[MT * LDA];
    __shared__ __attribute__((aligned(16))) float As1[DUAL ? MT * LDA : 4];

    const int tid  = threadIdx.x;
    const int row0 = blockIdx.x * MT;

    // cooperative A-tile load, float4-vectorized (rows clamped: keeps loads
    // in-bounds for the ragged last block while EXEC stays all-ones)
    for (int t = tid; t < MT * (HIDC / 4); t += NT) {
        const int r  = t >> 5;               // HIDC/4 = 32 float4 per row
        const int c4 = t & 31;
        int row = row0 + r;
        if (row >= N_NODES) row = N_NODES - 1;
        float4 v = ((const float4*)(A0 + (size_t)row * HIDC))[c4];
        if (rs0) {
            const float s = rs0[row];
            v.x *= s; v.y *= s; v.z *= s; v.w *= s;
        }
        *(float4*)&As0[r * LDA + c4 * 4] = v;
        if constexpr (DUAL)
            *(float4*)&As1[r * LDA + c4 * 4] =
                ((const float4*)(A1 + (size_t)row * HIDC))[c4];
    }
    __syncthreads();

    const int lane = tid & 31;
    const int lrow = lane & 15;
    const int kh   = lane >> 4;              // K-half selector (0 or 1)
    const int col  = ((tid >> 5) * 16) + lrow;

    v8f acc[MTILES];
#pragma unroll
    for (int mt = 0; mt < MTILES; ++mt) acc[mt] = (v8f){};

#pragma unroll 8
    for (int k0 = 0; k0 < HIDC; k0 += 4) {
        const int kk = k0 + 2 * kh;
        v2f b;
        b.x = W0[(size_t)kk * OUTC + col];
        b.y = W0[(size_t)(kk + 1) * OUTC + col];
#pragma unroll
        for (int mt = 0; mt < MTILES; ++mt) {
            v2f a;
            a.x = As0[(mt * 16 + lrow) * LDA + kk];
            a.y = As0[(mt * 16 + lrow) * LDA + kk + 1];
            acc[mt] = wmma_f32(a, b, acc[mt]);
        }
    }
    if constexpr (DUAL) {
#pragma unroll 8
        for (int k0 = 0; k0 < HIDC; k0 += 4) {
            const int kk = k0 + 2 * kh;
            v2f b;
            b.x = W1w[(size_t)kk * OUTC + col];
            b.y = W1w[(size_t)(kk + 1) * OUTC + col];
#pragma unroll
            for (int mt = 0; mt < MTILES; ++mt) {
                v2f a;
                a.x = As1[(mt * 16 + lrow) * LDA + kk];
                a.y = As1[(mt * 16 + lrow) * LDA + kk + 1];
                acc[mt] = wmma_f32(a, b, acc[mt]);
            }
        }
    }

    const float bv = bias ? bias[col] : 0.0f;
#pragma unroll
    for (int mt = 0; mt < MTILES; ++mt) {
#pragma unroll
        for (int r = 0; r < 8; ++r) {
            const int row = row0 + mt * 16 + r + 8 * kh;
            if (row < N_NODES)               // divergence only after all WMMAs
                C[(size_t)row * OUTC + col] = apply_act<ACT>(acc[mt][r] + bv);
        }
    }
}

// ---------------------------------------------------------------------------
// Degree count (float atomics; deg excludes self-loop)
// ---------------------------------------------------------------------------
__global__ void deg_kernel(const long long* __restrict__ dst, float* __restrict__ deg) {
    const int e = blockIdx.x * blockDim.x + threadIdx.x;
    if (e >= N_EDGES) return;
    atomicAdd(&deg[dst[e]], 1.0f);
}

__global__ void norm_kernel(const float* __restrict__ deg,
                            float* __restrict__ dinv, float* __restrict__ cntinv) {
    const int n = blockIdx.x * blockDim.x + threadIdx.x;
    if (n >= N_NODES) return;
    const float d = deg[n];
    dinv[n]   = rsqrtf(d + 1.0f);           // GCN deg includes self-loop -> always > 0
    cntinv[n] = 1.0f / fmaxf(d, 1.0f);      // SAGE mean divisor
}

// ---------------------------------------------------------------------------
// GCN edge scatter: agg[d] += X[s] * dinv[s]*dinv[d].  1 wave / edge,
// each lane owns 4 contiguous channels (float4 load + 4 atomic f32 adds).
// ---------------------------------------------------------------------------
__global__ void gcn_scatter(const float* __restrict__ X,
                            const long long* __restrict__ src,
                            const long long* __restrict__ dst,
                            const float* __restrict__ dinv,
                            float* __restrict__ agg) {
    const int gid = blockIdx.x * blockDim.x + threadIdx.x;
    const int e = gid >> 5, lane = gid & 31;
    if (e >= N_EDGES) return;
    const long long s = src[e], d = dst[e];
    const float w = dinv[s] * dinv[d];
    const float4 xv = ((const float4*)(X + (size_t)s * HIDC))[lane];
    float* o = agg + (size_t)d * HIDC + lane * 4;
    atomicAdd(o + 0, xv.x * w);
    atomicAdd(o + 1, xv.y * w);
    atomicAdd(o + 2, xv.z * w);
    atomicAdd(o + 3, xv.w * w);
}

// h1 = relu(agg + X*dinv^2 + b_gcn)   (self-loop term folded in, atomic-free)
__global__ void gcn_epilogue(const float* __restrict__ agg, const float* __restrict__ X,
                             const float* __restrict__ dinv, const float* __restrict__ b,
                             float* __restrict__ h1) {
    const int gid = blockIdx.x * blockDim.x + threadIdx.x;
    if (gid >= N_NODES * HIDC) return;
    const int node = gid >> 7, c = gid & (HIDC - 1);
    const float di = dinv[node];
    h1[gid] = fmaxf(agg[gid] + X[gid] * di * di + b[c], 0.0f);
}

// SAGE edge scatter: neigh[d] += h1[s]
__global__ void sage_scatter(const float* __restrict__ h1,
                             const long long* __restrict__ src,
                             const long long* __restrict__ dst,
                             float* __restrict__ neigh) {
    const int gid = blockIdx.x * blockDim.x + threadIdx.x;
    const int e = gid >> 5, lane = gid & 31;
    if (e >= N_EDGES) return;
    const long long s = src[e], d = dst[e];
    const float4 hv = ((const float4*)(h1 + (size_t)s * HIDC))[lane];
    float* o = neigh + (size_t)d * HIDC + lane * 4;
    atomicAdd(o + 0, hv.x);
    atomicAdd(o + 1, hv.y);
    atomicAdd(o + 2, hv.z);
    atomicAdd(o + 3, hv.w);
}

// values[n] = dot(h2[n,:], Wv) + bv   -- one wave per node, shuffle reduction
__global__ void value_kernel(const float* __restrict__ h2, const float* __restrict__ Wv,
                             const float* __restrict__ bv, float* __restrict__ out) {
    const int gid = blockIdx.x * blockDim.x + threadIdx.x;
    const int node = gid >> 5, lane = gid & 31;
    if (node >= N_NODES) return;
    const float4 hv = ((const float4*)(h2 + (size_t)node * HIDC))[lane];
    const float4 wv = ((const float4*)Wv)[lane];
    float p = hv.x * wv.x + hv.y * wv.y + hv.z * wv.z + hv.w * wv.w;
#pragma unroll
    for (int off = 16; off > 0; off >>= 1) p += __shfl_xor(p, off, 32);
    if (lane == 0) out[node] = p + bv[0];
}

// ---------------------------------------------------------------------------
extern "C" void kernel_launch(void* const* d_in, const int* in_sizes, int n_in,
                              void* d_out, int out_size, void* d_ws, size_t ws_size,
                              hipStream_t stream) {
    const float*     feat     = (const float*)d_in[0];
    const long long* ei       = (const long long*)d_in[1];   // int64 edge_index [2,E]
    const float*     W_gcn    = (const float*)d_in[2];
    const float*     b_gcn    = (const float*)d_in[3];
    const float*     W_sage_l = (const float*)d_in[4];
    const float*     b_sage_l = (const float*)d_in[5];
    const float*     W_sage_r = (const float*)d_in[6];
    const float*     W1       = (const float*)d_in[7];
    const float*     b1       = (const float*)d_in[8];
    const float*     W2       = (const float*)d_in[9];
    const float*     b2       = (const float*)d_in[10];
    const float*     W3       = (const float*)d_in[11];
    const float*     b3       = (const float*)d_in[12];
    const float*     Wv       = (const float*)d_in[13];
    const float*     bv       = (const float*)d_in[14];

    float* out_means = (float*)d_out;                           // [N, 64]
    float* out_vals  = (float*)d_out + (size_t)N_NODES * OUTD;  // [N]

    char* ws = (char*)d_ws;
    const size_t BUF = (size_t)N_NODES * HIDC * sizeof(float);  // 10.24 MB
    float* bX   = (float*)(ws + 0 * BUF);   // X, later reused as SAGE neigh-sum
    float* bAgg = (float*)(ws + 1 * BUF);   // GCN accumulator, later z1
    float* bH1  = (float*)(ws + 2 * BUF);   // h1, later z2
    float* bH2  = (float*)(ws + 3 * BUF);   // h2
    float* deg    = (float*)(ws + 4 * BUF);
    float* dinv   = deg + N_NODES;
    float* cntinv = dinv + N_NODES;

    const long long* src = ei;
    const long long* dst = ei + N_EDGES;

    // zero accumulators (required every launch: harness does not re-clear ws)
    hipMemsetAsync(deg, 0, N_NODES * sizeof(float), stream);
    hipMemsetAsync(bAgg, 0, BUF, stream);

    deg_kernel<<<(N_EDGES + 255) / 256, 256, 0, stream>>>(dst, deg);
    norm_kernel<<<(N_NODES + 255) / 256, 256, 0, stream>>>(deg, dinv, cntinv);

    // grid sizes: single-A GEMMs use 64-row blocks, dual-A uses 32-row blocks
    const int g64 = (N_NODES + 63) / 64;    // 313 (last block row-guarded)
    const int g32 = (N_NODES + 31) / 32;    // 625 (exact)

    // X = feat @ W_gcn
    gemm_wmma<HIDC, 0, false, 4><<<g64, 256, 0, stream>>>(
        feat, nullptr, W_gcn, nullptr, nullptr, nullptr, bX);

    // GCN aggregate + epilogue -> h1
    gcn_scatter<<<(N_EDGES * 32) / 256, 256, 0, stream>>>(bX, src, dst, dinv, bAgg);
    gcn_epilogue<<<(N_NODES * HIDC + 255) / 256, 256, 0, stream>>>(bAgg, bX, dinv, b_gcn, bH1);

    // SAGE aggregate (reuse bX as neighbor sum)
    hipMemsetAsync(bX, 0, BUF, stream);
    sage_scatter<<<(N_EDGES * 32) / 256, 256, 0, stream>>>(bH1, src, dst, bX);

    // h2 = relu((neigh*cntinv) @ W_sage_l + h1 @ W_sage_r + b_sage_l)
    gemm_wmma<HIDC, 1, true, 2><<<g32, 256, 0, stream>>>(
        bX, cntinv, W_sage_l, bH1, W_sage_r, b_sage_l, bH2);

    // policy MLP
    gemm_wmma<HIDC, 2, false, 4><<<g64, 256, 0, stream>>>(
        bH2, nullptr, W1, nullptr, nullptr, b1, bAgg);             // z1
    gemm_wmma<HIDC, 2, false, 4><<<g64, 256, 0, stream>>>(
        bAgg, nullptr, W2, nullptr, nullptr, b2, bH1);             // z2
    gemm_wmma<OUTD, 0, false, 4><<<g64, 128, 0, stream>>>(
        bH1, nullptr, W3, nullptr, nullptr, b3, out_means);        // means

    // value head
    value_kernel<<<(N_NODES * 32 + 255) / 256, 256, 0, stream>>>(bH2, Wv, bv, out_vals);
}